// rAdjGCN_10075993276648
// MI455X (gfx1250) — compile-verified
//
#include <hip/hip_runtime.h>
#include <stdint.h>

#define N_USERS  100000
#define M_ITEMS  50000
#define N_NODES  (N_USERS + M_ITEMS)
#define DIM      128
#define N_INTER  1000000
#define NF       (N_NODES * DIM)   /* 19,200,000 floats */
#define NF4      (NF / 4)          /* 4,800,000 float4 */

// ---------------------------------------------------------------------------
// CDNA5 async global->LDS helpers (inline asm: portable across toolchains)
// ---------------------------------------------------------------------------
__device__ __forceinline__ void async_row_b128(uint32_t lds_byte,
                                               const float* __restrict__ gsrc,
                                               int lane) {
  // 32 lanes x b128 = one 512B feature row lands contiguously in LDS
  uint64_t ga = (uint64_t)(uintptr_t)(gsrc + (lane << 2));
  uint32_t la = lds_byte + ((uint32_t)lane << 4);
  asm volatile("global_load_async_to_lds_b128 %0, %1, off"
               :: "v"(la), "v"(ga) : "memory");
}
__device__ __forceinline__ void wait_async_le2() {
  asm volatile("s_wait_asynccnt 0x2" ::: "memory");
}
__device__ __forceinline__ void wait_async_0() {
  asm volatile("s_wait_asynccnt 0x0" ::: "memory");
}
__device__ __forceinline__ void wait_ds0() {
  // WAR safety: prior ds reads of the recycled buffer retired before the
  // async engine can overwrite it
  asm volatile("s_wait_dscnt 0x0" ::: "memory");
}

// ---------------------------------------------------------------------------
// Setup kernels
// ---------------------------------------------------------------------------
__global__ void k_init(const float4* __restrict__ x, float4* __restrict__ out,
                       float4* __restrict__ ha, float4* __restrict__ hb, int n4) {
  int i = blockIdx.x * blockDim.x + threadIdx.x;
  if (i < n4) {
    float4 v = x[i];
    out[i] = v;
    ha[i]  = v;
    float4 z; z.x = 0.f; z.y = 0.f; z.z = 0.f; z.w = 0.f;
    hb[i]  = z;
  }
}

__global__ void k_zero_deg(unsigned int* __restrict__ deg) {
  int i = blockIdx.x * blockDim.x + threadIdx.x;
  if (i < N_NODES) deg[i] = 0u;
}

__global__ void k_deg(const int* __restrict__ user, const int* __restrict__ item,
                      unsigned int* __restrict__ deg) {
  int e = blockIdx.x * blockDim.x + threadIdx.x;
  if (e < N_INTER) {
    atomicAdd(&deg[user[e]], 1u);
    atomicAdd(&deg[item[e] + N_USERS], 1u);
  }
}

__global__ void k_invs(const unsigned int* __restrict__ deg,
                       float* __restrict__ invs) {
  int i = blockIdx.x * blockDim.x + threadIdx.x;
  if (i < N_NODES) {
    float d = (float)deg[i];
    if (d <= 0.f) d = 1e-6f;          // reference's zero-degree clamp
    invs[i] = 1.0f / sqrtf(d);
  }
}

// per-interaction symmetric weight: same for both edge directions
__global__ void k_edgew(const int* __restrict__ user, const int* __restrict__ item,
                        const float* __restrict__ invs, float* __restrict__ ew) {
  int k = blockIdx.x * blockDim.x + threadIdx.x;
  if (k < N_INTER) ew[k] = invs[user[k]] * invs[item[k] + N_USERS];
}

// ---------------------------------------------------------------------------
// Interaction scatter: one wave per interaction (both directions at once).
// Async-pipelined double-buffered gather through LDS (2 rows / stage),
// coalesced fp32 atomics into hout (hout must be pre-zeroed).
// ---------------------------------------------------------------------------
__global__ __launch_bounds__(256)
void k_scatter(const float* __restrict__ hin, float* __restrict__ hout,
               const int* __restrict__ user, const int* __restrict__ item,
               const float* __restrict__ ew) {
  // 8 waves/block * 2 stages * 2 rows * 512B = 16KB
  __shared__ __align__(1024) float smem[8 * 512];
  const int lane = threadIdx.x & 31;
  const int wib  = threadIdx.x >> 5;
  const int wave = blockIdx.x * 8 + wib;
  const int nwav = gridDim.x * 8;
  float* mybuf = smem + wib * 512;
  const uint32_t base = (uint32_t)(uintptr_t)mybuf;

  int k = wave;
  if (k >= N_INTER) return;

  auto issue_pair = [&](int p, int u, int it) {
    uint32_t off = base + (uint32_t)p * 1024u;
    async_row_b128(off,        hin + (size_t)u  * DIM, lane);
    async_row_b128(off + 512u, hin + (size_t)it * DIM, lane);
  };
  auto process_pair = [&](int p, int u, int it, float wt) {
    const float* bu = mybuf + p * 256;     // gathered user row
    const float* bi = bu + 128;            // gathered item row
    // stride-32 LDS reads: conflict-free across 64 banks
    float u0 = bu[lane], u1 = bu[lane + 32], u2 = bu[lane + 64], u3 = bu[lane + 96];
    float i0 = bi[lane], i1 = bi[lane + 32], i2 = bi[lane + 64], i3 = bi[lane + 96];
    float* du = hout + (size_t)it * DIM + lane;   // user row -> item node
    float* di = hout + (size_t)u  * DIM + lane;   // item row -> user node
    // each atomic instruction covers 32 consecutive floats across the wave
    atomicAdd(du,      u0 * wt); atomicAdd(du + 32, u1 * wt);
    atomicAdd(du + 64, u2 * wt); atomicAdd(du + 96, u3 * wt);
    atomicAdd(di,      i0 * wt); atomicAdd(di + 32, i1 * wt);
    atomicAdd(di + 64, i2 * wt); atomicAdd(di + 96, i3 * wt);
  };

  int   u  = user[k];
  int   it = item[k] + N_USERS;
  float wt = ew[k];
  issue_pair(0, u, it);
  int p = 0;

  for (int kn = k + nwav; kn < N_INTER; kn += nwav) {
    int   u2  = user[kn];
    int   it2 = item[kn] + N_USERS;
    float wt2 = ew[kn];
    wait_ds0();                        // recycled buffer's reads retired
    issue_pair(p ^ 1, u2, it2);        // prefetch next interaction's rows
    wait_async_le2();                  // current stage's 2 rows have arrived
    process_pair(p, u, it, wt);
    p ^= 1; u = u2; it = it2; wt = wt2;
  }

  wait_async_0();
  process_pair(p, u, it, wt);
}

// ---------------------------------------------------------------------------
// out = (out + hadd) * scale ; optionally zero the next scatter target
// ---------------------------------------------------------------------------
__global__ __launch_bounds__(256)
void k_accum(float4* __restrict__ out, const float4* __restrict__ hadd,
             float4* __restrict__ hzero, float scale, int n4) {
  int i = blockIdx.x * blockDim.x + threadIdx.x;
  if (i < n4) {
    float4 o = out[i];
    float4 h = hadd[i];
    o.x = (o.x + h.x) * scale;
    o.y = (o.y + h.y) * scale;
    o.z = (o.z + h.z) * scale;
    o.w = (o.w + h.w) * scale;
    out[i] = o;
    if (hzero) {
      float4 z; z.x = 0.f; z.y = 0.f; z.z = 0.f; z.w = 0.f;
      hzero[i] = z;
    }
  }
}

// ---------------------------------------------------------------------------
extern "C" void kernel_launch(void* const* d_in, const int* in_sizes, int n_in,
                              void* d_out, int out_size, void* d_ws, size_t ws_size,
                              hipStream_t stream) {
  (void)in_sizes; (void)n_in; (void)out_size; (void)ws_size;

  const float* x    = (const float*)d_in[0];
  const int*   user = (const int*)d_in[1];
  const int*   item = (const int*)d_in[2];
  float*       out  = (float*)d_out;

  float*        ha   = (float*)d_ws;             // 76.8 MB
  float*        hb   = ha + NF;                  // 76.8 MB
  unsigned int* deg  = (unsigned int*)(hb + NF); // 600 KB
  float*        invs = (float*)(deg + N_NODES);  // 600 KB
  float*        ewt  = invs + N_NODES;           // 4 MB

  const int T = 256;
  const int gF  = (NF4 + T - 1) / T;
  const int gN  = (N_NODES + T - 1) / T;
  const int gE  = (N_INTER + T - 1) / T;
  const int gSc = 2048;                          // 16384 waves, ~61 stages each

  k_zero_deg<<<gN, T, 0, stream>>>(deg);
  k_init<<<gF, T, 0, stream>>>((const float4*)x, (float4*)out, (float4*)ha,
                               (float4*)hb, NF4);
  k_deg<<<gE, T, 0, stream>>>(user, item, deg);
  k_invs<<<gN, T, 0, stream>>>(deg, invs);
  k_edgew<<<gE, T, 0, stream>>>(user, item, invs, ewt);

  // layer 1: h1 = Â x   (ha -> hb), out += h1, zero ha
  k_scatter<<<gSc, T, 0, stream>>>(ha, hb, user, item, ewt);
  k_accum<<<gF, T, 0, stream>>>((float4*)out, (const float4*)hb, (float4*)ha,
                                1.0f, NF4);
  // layer 2: h2 = Â h1  (hb -> ha), out += h2, zero hb
  k_scatter<<<gSc, T, 0, stream>>>(hb, ha, user, item, ewt);
  k_accum<<<gF, T, 0, stream>>>((float4*)out, (const float4*)ha, (float4*)hb,
                                1.0f, NF4);
  // layer 3: h3 = Â h2  (ha -> hb), out = (out + h3) / 4
  k_scatter<<<gSc, T, 0, stream>>>(ha, hb, user, item, ewt);
  k_accum<<<gF, T, 0, stream>>>((float4*)out, (const float4*)hb, (float4*)nullptr,
                                0.25f, NF4);
}